// MetaUpScale_37658273251414
// MI455X (gfx1250) — compile-verified
//
#include <hip/hip_runtime.h>

// Problem constants (fixed by the reference: H=W=128, C=64, K=3, HID=256, OUT_C=3, scale=2)
#define H 128
#define W 128
#define C 64
#define KS 3
#define HID 256
#define OUT_C 3
#define WSZ (KS*KS*C*OUT_C)   // 1728 weights per output pixel (incl. bias_3 folded in)
#define SCALE 2
#define OH (H*SCALE)          // 256
#define OW (W*SCALE)          // 256
#define NPHASE (SCALE*SCALE)  // 4 distinct MLP inputs -> 4 distinct weight tensors

typedef __attribute__((ext_vector_type(2))) float v2f;
typedef __attribute__((ext_vector_type(8))) float v8f;

// ---------------------------------------------------------------------------
// Kernel A1: layers 1+2 of the MLP for the 4 phases.
// One block of 256 threads; thread t owns hidden unit t.
// Output: h2 padded to 16 rows (rows 4..15 = 0) for the WMMA A-matrix.
// ---------------------------------------------------------------------------
__global__ __launch_bounds__(HID) void mlp_h2_kernel(
    const float* __restrict__ k1, const float* __restrict__ b1,
    const float* __restrict__ k2, const float* __restrict__ b2,
    float* __restrict__ h2_out /* 16 x HID */) {
  __shared__ float h1s[NPHASE][HID];
  const int t = threadIdx.x;

  // Layer 1: V(p) = [0.5*(p>>1), 0.5*(p&1), 0.5]  (frac parts for scale=2)
  const float b1v = b1[t];
  const float w0 = k1[0 * HID + t];
  const float w1 = k1[1 * HID + t];
  const float w2 = k1[2 * HID + t];
#pragma unroll
  for (int p = 0; p < NPHASE; ++p) {
    const float vi = 0.5f * (float)(p >> 1);
    const float vj = 0.5f * (float)(p & 1);
    const float h = fmaf(vi, w0, fmaf(vj, w1, fmaf(0.5f, w2, b1v)));
    h1s[p][t] = fmaxf(h, 0.0f);
  }
  __syncthreads();

  // Layer 2: 4 rows x 256x256 GEMM; k2 column reads are lane-coalesced,
  // h1s reads are lane-uniform LDS broadcasts.
  const float b2v = b2[t];
  float acc0 = b2v, acc1 = b2v, acc2 = b2v, acc3 = b2v;
#pragma unroll 4
  for (int k = 0; k < HID; ++k) {
    const float wv = k2[k * HID + t];
    acc0 = fmaf(h1s[0][k], wv, acc0);
    acc1 = fmaf(h1s[1][k], wv, acc1);
    acc2 = fmaf(h1s[2][k], wv, acc2);
    acc3 = fmaf(h1s[3][k], wv, acc3);
  }
  h2_out[0 * HID + t] = fmaxf(acc0, 0.0f);
  h2_out[1 * HID + t] = fmaxf(acc1, 0.0f);
  h2_out[2 * HID + t] = fmaxf(acc2, 0.0f);
  h2_out[3 * HID + t] = fmaxf(acc3, 0.0f);
#pragma unroll
  for (int p = NPHASE; p < 16; ++p) h2_out[p * HID + t] = 0.0f;  // WMMA pad rows
}

// ---------------------------------------------------------------------------
// Kernel A2: W(16x1728) = h2(16x256) @ k3(256x1728) + bias_3 via
// V_WMMA_F32_16X16X4_F32 (fp32-exact). One wave per 16-column tile, 64 K-steps.
// A layout (ISA 7.12.2, 32-bit A 16x4): lane&15 = M, lane>>4 selects K pair
//   {0,1} vs {2,3}; VGPR0/1 = the two K values.
// B layout (dual of A, 4x16): lane&15 = N, lane>>4 selects K pair.
// D layout: lanes 0-15 hold M=0..7 in VGPR0..7 (we need M=0..3 -> acc[0..3]).
// ---------------------------------------------------------------------------
__global__ __launch_bounds__(32) void mlp_w_wmma_kernel(
    const float* __restrict__ h2 /* 16 x HID */,
    const float* __restrict__ k3 /* HID x WSZ */,
    const float* __restrict__ b3 /* WSZ */,
    float* __restrict__ wout /* NPHASE x WSZ */) {
  const int ntile = blockIdx.x;        // 0..107 (1728/16)
  const int lane  = threadIdx.x;       // wave32
  const int m     = lane & 15;
  const int half  = lane >> 4;
  const int ncol  = ntile * 16 + m;

  v8f acc = {};
#pragma unroll 4
  for (int k0 = 0; k0 < HID / 4; ++k0) {
    const int ka = k0 * 4 + half * 2;
    v2f a, b;
    a.x = h2[m * HID + ka + 0];
    a.y = h2[m * HID + ka + 1];
    b.x = k3[(size_t)(ka + 0) * WSZ + ncol];
    b.y = k3[(size_t)(ka + 1) * WSZ + ncol];
    // 8 args: (neg_a, A, neg_b, B, c_mod, C, reuse_a, reuse_b)
    acc = __builtin_amdgcn_wmma_f32_16x16x4_f32(false, a, false, b,
                                                (short)0, acc, false, false);
  }
  if (lane < 16) {
    const float bias = b3[ncol];
#pragma unroll
    for (int p = 0; p < NPHASE; ++p)
      wout[p * WSZ + ncol] = acc[p] + bias;
  }
}

// ---------------------------------------------------------------------------
// Kernel B: per-pixel 3x3x64 window contraction with the phase weights.
// 32x8 output tile per block (256 threads = 8 wave32s).
//   Feature tile in LDS: 6 rows x 18 cols x 64 ch, C-stride padded to 65 so
//   the 16 distinct window base addresses per wave land in 16 distinct banks.
//   Weights in LDS: 4 x 1728 padded to stride 1744 (1744 % 64 = 16) so the
//   two phases active within a wave hit different banks (broadcast reads).
// ---------------------------------------------------------------------------
#define TW 32
#define TH 8
#define FROWS 6               // TH/2 + 2
#define FCOLS 18              // TW/2 + 2
#define CP 65                 // padded channel stride (64+1)
#define FROWSTR (FCOLS * CP)  // 1170
#define WSTR (WSZ + 16)       // 1744

__global__ __launch_bounds__(TW * TH) void upscale_kernel(
    const float* __restrict__ feat /* H x W x C */,
    const float* __restrict__ wphase /* NPHASE x WSZ */,
    float* __restrict__ out /* OH*OW x OUT_C */) {
  __shared__ float fs[FROWS * FROWSTR];   // 7020 floats = 28.1 KB
  __shared__ float ws[NPHASE * WSTR];     // 6976 floats = 27.9 KB

  const int tx = threadIdx.x, ty = threadIdx.y;
  const int tid = ty * TW + tx;
  const int j0 = blockIdx.x * TW;
  const int i0 = blockIdx.y * TH;
  const int bi = (i0 >> 1) - 1;  // top input row of tile (may be -1 -> zero pad)
  const int bj = (j0 >> 1) - 1;

  // Cooperative load of the input window tile (coalesced along C).
  for (int idx = tid; idx < FROWS * FCOLS * C; idx += TW * TH) {
    const int r   = idx / (FCOLS * C);
    const int rem = idx - r * (FCOLS * C);
    const int col = rem / C;
    const int c   = rem - col * C;
    const int gi = bi + r, gj = bj + col;
    float v = 0.0f;
    if (gi >= 0 && gi < H && gj >= 0 && gj < W)
      v = feat[((size_t)gi * W + gj) * C + c];
    fs[r * FROWSTR + col * CP + c] = v;
  }
  // Cooperative load of all 4 phase weight sets (L2-resident, read 256x).
  for (int idx = tid; idx < NPHASE * WSZ; idx += TW * TH) {
    const int p = idx / WSZ;
    const int o = idx - p * WSZ;
    ws[p * WSTR + o] = wphase[idx];
  }
  __syncthreads();

  const int phase = ((ty & 1) << 1) | (tx & 1);
  const float* wp = &ws[phase * WSTR];
  const int rbase = ty >> 1;
  const int cbase = tx >> 1;

  float a0 = 0.0f, a1 = 0.0f, a2 = 0.0f;
#pragma unroll
  for (int p = 0; p < KS; ++p) {
    const float* frow = &fs[(rbase + p) * FROWSTR];
#pragma unroll
    for (int q = 0; q < KS; ++q) {
      const float* f  = &frow[(cbase + q) * CP];
      const float* wv = &wp[(p * KS + q) * C * OUT_C];
#pragma unroll 8
      for (int c = 0; c < C; ++c) {
        const float fv = f[c];
        a0 = fmaf(fv, wv[c * 3 + 0], a0);
        a1 = fmaf(fv, wv[c * 3 + 1], a1);
        a2 = fmaf(fv, wv[c * 3 + 2], a2);
      }
    }
  }

  const int i = i0 + ty, j = j0 + tx;
  const size_t o = ((size_t)i * OW + j) * OUT_C;
  out[o + 0] = a0;
  out[o + 1] = a1;
  out[o + 2] = a2;
}

// ---------------------------------------------------------------------------
extern "C" void kernel_launch(void* const* d_in, const int* in_sizes, int n_in,
                              void* d_out, int out_size, void* d_ws, size_t ws_size,
                              hipStream_t stream) {
  const float* feat = (const float*)d_in[0];
  const float* k1   = (const float*)d_in[1];
  const float* b1   = (const float*)d_in[2];
  const float* k2   = (const float*)d_in[3];
  const float* b2   = (const float*)d_in[4];
  const float* k3   = (const float*)d_in[5];
  const float* b3   = (const float*)d_in[6];
  // d_in[7] = scale (==2, baked into the launch geometry)

  float* h2   = (float*)d_ws;        // 16 x 256 floats
  float* wout = h2 + 16 * HID;       // 4 x 1728 floats (total ws use: 44 KB)

  mlp_h2_kernel<<<1, HID, 0, stream>>>(k1, b1, k2, b2, h2);
  mlp_w_wmma_kernel<<<WSZ / 16, 32, 0, stream>>>(h2, k3, b3, wout);

  dim3 blk(TW, TH);
  dim3 grd(OW / TW, OH / TH);
  upscale_kernel<<<grd, blk, 0, stream>>>(feat, wout, (float*)d_out);
}